// TemporalAttention_1460288880725
// MI455X (gfx1250) — compile-verified
//
#include <hip/hip_runtime.h>
#include <hip/hip_bf16.h>
#include <math.h>

// ---------------------------------------------------------------------------
// TemporalAttention, algebraically reduced:
//   q_last = x[:,S-1,:] @ Wq.T + bq                       (32x1024 GEMM, WMMA f32)
//   qk     = q_last @ Wk            qbk[b] = q_last[b].bk (32x1024 GEMM, WMMA f32)
//   scores[b,s] = (x[b,s,:].qk[b,:] + qbk[b]) / 32        (stream x, pass 1)
//   weights = softmax(scores)                              -> d_out[32768..]
//   xc[b,:] = sum_s weights[b,s] * x[b,s,:]               (stream x, pass 2)
//   context = xc @ Wv.T + bv                               -> d_out[0..32768)
// ---------------------------------------------------------------------------

#define HID 1024
#define BAT 32
#define SEQ 2048

typedef __attribute__((ext_vector_type(2))) float v2f;
typedef __attribute__((ext_vector_type(8))) float v8f;

// ---- small GEMM: Y[32 x 1024] = A[32 x 1024] * B + bias, via V_WMMA_F32_16X16X4_F32
// BT=true : B(k,n) = W[n*1024 + k]   (y = a @ W.T, torch-Linear layout)
// BT=false: B(k,n) = W[k*1024 + n]   (y = a @ W)
// A row m lives at A + m*astride (lets us use strided x[:,S-1,:] directly).
template <bool BT>
__global__ __launch_bounds__(128) void gemm32_wmma(
    const float* __restrict__ A, long astride,
    const float* __restrict__ W,
    const float* __restrict__ bias,
    float* __restrict__ Y) {
  const int wave  = (int)((blockIdx.x * blockDim.x + threadIdx.x) >> 5); // 0..127
  const int lane  = (int)(threadIdx.x & 31);
  const int mtile = wave >> 6;   // 0..1   (BAT/16)
  const int ntile = wave & 63;   // 0..63  (HID/16)
  const int l15   = lane & 15;
  const int hi    = lane >> 4;               // 0: lanes 0-15, 1: lanes 16-31
  const int m     = mtile * 16 + l15;        // A row for this lane
  const int n     = ntile * 16 + l15;        // B col for this lane
  const float* __restrict__ Arow = A + (long)m * astride;

  v8f c = {};
#pragma unroll 4
  for (int k = 0; k < HID; k += 4) {
    // A fragment: VGPR0/1 hold K={0,1} for lanes 0-15, K={2,3} for lanes 16-31
    float4 av = *(const float4*)(Arow + k);
    v2f a;
    a.x = hi ? av.z : av.x;
    a.y = hi ? av.w : av.y;
    // B fragment: mirrored layout over N
    v2f b;
    if (BT) {
      float4 bv4 = *(const float4*)(W + (long)n * HID + k);
      b.x = hi ? bv4.z : bv4.x;
      b.y = hi ? bv4.w : bv4.y;
    } else {
      const int kk = k + hi * 2;
      b.x = W[(long)(kk + 0) * HID + n];
      b.y = W[(long)(kk + 1) * HID + n];
    }
    c = __builtin_amdgcn_wmma_f32_16x16x4_f32(
        /*neg_a=*/false, a, /*neg_b=*/false, b,
        /*c_mod=*/(short)0, c, /*reuse_a=*/false, /*reuse_b=*/false);
  }

  const float bcol = bias ? bias[n] : 0.0f;
  const int row0 = mtile * 16 + hi * 8;  // lanes 0-15 -> M=v, lanes 16-31 -> M=8+v
#pragma unroll
  for (int v = 0; v < 8; ++v)
    Y[(long)(row0 + v) * HID + n] = c[v] + bcol;
}

// ---- qbk[b] = dot(q_last[b,:], bk)  -- one wave per batch
__global__ __launch_bounds__(256) void qbk_kernel(
    const float* __restrict__ qlast, const float* __restrict__ bk,
    float* __restrict__ qbk) {
  const int b    = (int)((blockIdx.x * 256 + threadIdx.x) >> 5); // 0..31
  const int lane = (int)(threadIdx.x & 31);
  const float4* qr = (const float4*)(qlast + (long)b * HID);
  const float4* br = (const float4*)bk;
  float acc = 0.f;
#pragma unroll
  for (int j = 0; j < 8; ++j) {
    float4 q = qr[j * 32 + lane];
    float4 v = br[j * 32 + lane];
    acc = fmaf(q.x, v.x, acc); acc = fmaf(q.y, v.y, acc);
    acc = fmaf(q.z, v.z, acc); acc = fmaf(q.w, v.w, acc);
  }
  for (int off = 16; off > 0; off >>= 1) acc += __shfl_xor(acc, off, 32);
  if (lane == 0) qbk[b] = acc;
}

// ---- scores[b,s] = (x[b,s,:].qk[b,:] + qbk[b]) / 32 -- one wave per (b,s), streams x
__global__ __launch_bounds__(256) void scores_kernel(
    const float* __restrict__ x, const float* __restrict__ qk,
    const float* __restrict__ qbk, float* __restrict__ scores) {
  const int gw   = (int)((blockIdx.x * 256 + threadIdx.x) >> 5);
  const int lane = (int)(threadIdx.x & 31);
  const int b = gw >> 11;          // /SEQ
  const int s = gw & (SEQ - 1);
  const float4* xr = (const float4*)(x + ((size_t)b * SEQ + s) * HID);
  const float4* qr = (const float4*)(qk + (long)b * HID);
  float acc = 0.f;
#pragma unroll
  for (int j = 0; j < 8; ++j) {
    float4 xv = xr[j * 32 + lane];
    float4 qv = qr[j * 32 + lane];
    acc = fmaf(xv.x, qv.x, acc); acc = fmaf(xv.y, qv.y, acc);
    acc = fmaf(xv.z, qv.z, acc); acc = fmaf(xv.w, qv.w, acc);
  }
  for (int off = 16; off > 0; off >>= 1) acc += __shfl_xor(acc, off, 32);
  if (lane == 0) scores[(long)b * SEQ + s] = (acc + qbk[b]) * (1.0f / 32.0f);
}

// ---- softmax over S per batch -- one block per batch, deterministic LDS trees
__global__ __launch_bounds__(256) void softmax_kernel(
    const float* __restrict__ scores, float* __restrict__ weights) {
  const int b = (int)blockIdx.x;
  const int t = (int)threadIdx.x;
  __shared__ float red[256];
  float v[8];
  float m = -INFINITY;
#pragma unroll
  for (int j = 0; j < 8; ++j) {
    v[j] = scores[(long)b * SEQ + j * 256 + t];
    m = fmaxf(m, v[j]);
  }
  red[t] = m; __syncthreads();
  for (int off = 128; off > 0; off >>= 1) {
    if (t < off) red[t] = fmaxf(red[t], red[t + off]);
    __syncthreads();
  }
  m = red[0]; __syncthreads();
  float sum = 0.f;
#pragma unroll
  for (int j = 0; j < 8; ++j) { v[j] = expf(v[j] - m); sum += v[j]; }
  red[t] = sum; __syncthreads();
  for (int off = 128; off > 0; off >>= 1) {
    if (t < off) red[t] += red[t + off];
    __syncthreads();
  }
  const float inv = 1.0f / red[0];
#pragma unroll
  for (int j = 0; j < 8; ++j)
    weights[(long)b * SEQ + j * 256 + t] = v[j] * inv;
}

// ---- xc partials: part[b][ch][h] = sum_{s in chunk ch} w[b,s]*x[b,s,h]
// grid 32*16 blocks, each streams a 512 KiB slab of x; float4 per thread (256 thr = H).
__global__ __launch_bounds__(256) void xc_part_kernel(
    const float* __restrict__ x, const float* __restrict__ weights,
    float* __restrict__ part) {
  const int CH = 16, SPC = SEQ / CH;   // 128 s per chunk
  const int b  = (int)(blockIdx.x >> 4);
  const int ch = (int)(blockIdx.x & 15);
  const int t  = (int)threadIdx.x;
  const float4* xb = (const float4*)(x + ((size_t)b * SEQ + (size_t)ch * SPC) * HID);
  const float* wb  = weights + (long)b * SEQ + ch * SPC;
  float4 acc = {0.f, 0.f, 0.f, 0.f};
  for (int s = 0; s < SPC; ++s) {
    const float w = wb[s];
    float4 xv = xb[(long)s * 256 + t];
    acc.x = fmaf(w, xv.x, acc.x); acc.y = fmaf(w, xv.y, acc.y);
    acc.z = fmaf(w, xv.z, acc.z); acc.w = fmaf(w, xv.w, acc.w);
  }
  ((float4*)part)[((long)b * 16 + ch) * 256 + t] = acc;
}

// ---- xc[b,h] = sum over 16 chunks (deterministic, no atomics)
__global__ __launch_bounds__(256) void xc_reduce_kernel(
    const float* __restrict__ part, float* __restrict__ xc) {
  const int i = (int)(blockIdx.x * 256 + threadIdx.x);  // 0..32767
  const int b = i >> 10, h = i & 1023;
  float s = 0.f;
#pragma unroll
  for (int c = 0; c < 16; ++c) s += part[((long)b * 16 + c) * HID + h];
  xc[i] = s;
}

extern "C" void kernel_launch(void* const* d_in, const int* in_sizes, int n_in,
                              void* d_out, int out_size, void* d_ws, size_t ws_size,
                              hipStream_t stream) {
  const float* x  = (const float*)d_in[0];
  const float* Wq = (const float*)d_in[1];
  const float* bq = (const float*)d_in[2];
  const float* Wk = (const float*)d_in[3];
  const float* bk = (const float*)d_in[4];
  const float* Wv = (const float*)d_in[5];
  const float* bv = (const float*)d_in[6];

  float* out     = (float*)d_out;
  float* context = out;                 // 32*1024
  float* weights = out + BAT * HID;     // 32*2048

  // workspace layout (floats, 16B-aligned offsets)
  float* ws     = (float*)d_ws;
  float* qlast  = ws;                    // 32768
  float* qk     = ws + 32768;            // 32768
  float* qbk    = ws + 65536;            // 32 (padded to 64)
  float* scores = ws + 65600;            // 65536
  float* xc     = ws + 131136;           // 32768
  float* part   = ws + 163904;           // 32*16*1024 = 524288

  // 1) q_last = x[:,S-1,:] @ Wq.T + bq   (A rows strided through x)
  gemm32_wmma<true><<<32, 128, 0, stream>>>(
      x + (size_t)(SEQ - 1) * HID, (long)SEQ * HID, Wq, bq, qlast);

  // 2) qk = q_last @ Wk ; qbk = q_last . bk
  gemm32_wmma<false><<<32, 128, 0, stream>>>(qlast, (long)HID, Wk, nullptr, qk);
  qbk_kernel<<<4, 256, 0, stream>>>(qlast, bk, qbk);

  // 3) scores (pass 1 over x): one wave per (b,s)
  scores_kernel<<<(BAT * SEQ) / 8, 256, 0, stream>>>(x, qk, qbk, scores);

  // 4) weights = softmax(scores)  -> d_out tail
  softmax_kernel<<<BAT, 256, 0, stream>>>(scores, weights);

  // 5) xc = weights^T x (pass 2 over x), chunked partials + deterministic reduce
  xc_part_kernel<<<BAT * 16, 256, 0, stream>>>(x, weights, part);
  xc_reduce_kernel<<<(BAT * HID) / 256, 256, 0, stream>>>(part, xc);

  // 6) context = xc @ Wv.T + bv -> d_out head
  gemm32_wmma<true><<<32, 128, 0, stream>>>(xc, (long)HID, Wv, bv, context);
}